// PerExampleGate_27693949124883
// MI455X (gfx1250) — compile-verified
//
#include <hip/hip_runtime.h>
#include <stdint.h>

typedef __attribute__((ext_vector_type(2))) float v2f;
typedef __attribute__((ext_vector_type(8))) float v8f;

#define B_ROWS      128            // rows of x per block
#define KC          64             // K-chunk staged in LDS
#define NCHUNKS     (P_DIM / KC)   // 16
#define LDS_STRIDE  68             // 64 + 4 pad floats (4-bank skew)
#define P_DIM       1024           // inner dimension
#define XTILE_FLT   (B_ROWS * LDS_STRIDE)          // 8704 floats
#define WTILE_FLT   (64 * LDS_STRIDE)              // 4352 floats
#define BUF_FLT     (XTILE_FLT + WTILE_FLT)        // 13056 floats = 52224 B
// GEMM cols: 0..7 gate logits, 8..55 expert gates, 56..63 pad (never read)

// Issue one K-chunk's global->LDS async copies: 8 instrs for the x tile (NT:
// x is streamed once, 256MB > L2, don't thrash the resident weights) + 4 for
// the weight tile (RT: L2-resident). Exactly 12 per wave, uniform across
// waves, so ASYNCcnt-based waits are exact. Pad weight rows 56..63 source a
// harmless valid address: they only feed output columns 56..63, never read.
__device__ __forceinline__ void issue_chunk_copies(
    const float* __restrict__ x, const float* __restrict__ Gw,
    const float* __restrict__ Ww, int rowBase, int kc,
    uint32_t ldsX, uint32_t ldsW, int tid)
{
    #pragma unroll
    for (int i = 0; i < 8; ++i) {                      // 2048 float4: 128 x 64
        const int f  = tid + i * 256;
        const int r  = f >> 4;                         // 16 float4 per row
        const int c4 = (f & 15) << 2;
        const unsigned long long ga =
            (unsigned long long)(const void*)(x + (size_t)(rowBase + r) * P_DIM + kc + c4);
        const uint32_t la = ldsX + (uint32_t)(r * LDS_STRIDE + c4) * 4u;
        asm volatile("global_load_async_to_lds_b128 %0, %1, off th:TH_LOAD_NT"
                     :: "v"(la), "v"(ga) : "memory");
    }
    #pragma unroll
    for (int i = 0; i < 4; ++i) {                      // 1024 float4: 64 x 64
        const int f  = tid + i * 256;
        const int j  = f >> 4;
        const int jc = (j < 56) ? j : 55;              // pad rows: don't-care
        const int c4 = (f & 15) << 2;
        const float* src = (jc < 8) ? (Gw + (size_t)jc * P_DIM)
                                    : (Ww + (size_t)(jc - 8) * P_DIM);
        const unsigned long long ga = (unsigned long long)(const void*)(src + kc + c4);
        const uint32_t la = ldsW + (uint32_t)(j * LDS_STRIDE + c4) * 4u;
        asm volatile("global_load_async_to_lds_b128 %0, %1, off"
                     :: "v"(la), "v"(ga) : "memory");
    }
}

__global__ __launch_bounds__(256) void dselect_fused_kernel(
    const float* __restrict__ x,
    const float* __restrict__ Gw,
    const float* __restrict__ Gb,
    const float* __restrict__ Ww,
    const float* __restrict__ Wb,
    float* __restrict__ out)
{
    extern __shared__ float smem[];
    // flat-aperture mapping: low 32 bits of a generic LDS pointer = LDS offset
    const uint32_t ldsBase = (uint32_t)(uintptr_t)smem;

    const int tid    = threadIdx.x;
    const int lane   = tid & 31;
    const int wave   = tid >> 5;
    const int laneLo = lane & 15;
    const int laneHi = lane >> 4;
    const int rowBase = blockIdx.x * B_ROWS;

    v8f acc[4];
    #pragma unroll
    for (int t = 0; t < 4; ++t)
        #pragma unroll
        for (int i = 0; i < 8; ++i) acc[t][i] = 0.0f;

    // ---- double-buffered async pipeline: copy chunk c+1 while WMMA-ing chunk c
    issue_chunk_copies(x, Gw, Ww, rowBase, 0,
                       ldsBase, ldsBase + XTILE_FLT * 4u, tid);

    for (int c = 0; c < NCHUNKS; ++c) {
        const int buf = c & 1;
        if (c + 1 < NCHUNKS) {
            const uint32_t off = (uint32_t)((buf ^ 1) * BUF_FLT) * 4u;
            issue_chunk_copies(x, Gw, Ww, rowBase, (c + 1) * KC,
                               ldsBase + off, ldsBase + off + XTILE_FLT * 4u, tid);
            asm volatile("s_wait_asynccnt 12" ::: "memory");   // chunk c copies done
        } else {
            asm volatile("s_wait_asynccnt 0" ::: "memory");
        }
        __syncthreads();   // chunk c visible to every wave

        const float* sXb = smem + buf * BUF_FLT;
        const float* sWb = sXb + XTILE_FLT;

        // --- WMMA f32 16x16x4: wave computes its 16 rows x 64 cols ---
        // A frag (16x4 f32): lane L: M = L&15, VGPR v holds K = (L>>4)*2 + v
        const float* ax = sXb + (wave * 16 + laneLo) * LDS_STRIDE + (laneHi << 1);
        #pragma unroll
        for (int ks = 0; ks < KC; ks += 4) {
            const v2f a = *(const v2f*)(ax + ks);
            #pragma unroll
            for (int t = 0; t < 4; ++t) {
                // B frag (4x16 f32): lane L: N = L&15, VGPR v holds K = (L>>4)*2 + v
                const v2f b = *(const v2f*)(sWb + (t * 16 + laneLo) * LDS_STRIDE
                                            + ks + (laneHi << 1));
                acc[t] = __builtin_amdgcn_wmma_f32_16x16x4_f32(
                    false, a, false, b, (short)0, acc[t], false, false);
            }
        }
        __syncthreads();   // all waves done reading buf; free for re-fill at c+2
    }

    // ---- dump P tile to LDS (reusing buffer-0 x-tile space; no pending writes)
    float* sP = smem;
    #pragma unroll
    for (int t = 0; t < 4; ++t)
        #pragma unroll
        for (int vv = 0; vv < 8; ++vv)
            sP[(wave * 16 + laneHi * 8 + vv) * LDS_STRIDE + t * 16 + laneLo] = acc[t][vv];
    __syncthreads();

    // ---- fused epilogue: each thread handles half a row (32 of 64 experts)
    const int r = tid >> 1;    // 0..127
    const int h = tid & 1;     // expert-index bit 5
    const float* pr = sP + r * LDS_STRIDE;

    // softmax over the 8 gating logits
    float l[8];
    #pragma unroll
    for (int c = 0; c < 8; ++c) l[c] = pr[c] + Gb[c];
    float mx = l[0];
    #pragma unroll
    for (int c = 1; c < 8; ++c) mx = fmaxf(mx, l[c]);
    float w[8], ssum = 0.f;
    #pragma unroll
    for (int c = 0; c < 8; ++c) { w[c] = __expf(l[c] - mx); ssum += w[c]; }
    const float inv = 1.0f / ssum;

    float outv[32];
    #pragma unroll
    for (int n = 0; n < 32; ++n) outv[n] = 0.f;

    #pragma unroll
    for (int e = 0; e < 8; ++e) {
        // smooth_step (gamma = 0.5): z = tc*(3 - 16*tc^2) + 0.5, tc = clip(g, -.25, .25)
        float z[6];
        #pragma unroll
        for (int j = 0; j < 6; ++j) {
            const float g  = pr[8 + e * 6 + j] + Wb[e * 6 + j];
            const float tc = fminf(fmaxf(g, -0.25f), 0.25f);
            z[j] = fmaf(tc, fmaf(-16.f, tc * tc, 3.f), 0.5f);
        }
        const float o0 = 1.f - z[0], o1 = 1.f - z[1], o2 = 1.f - z[2];
        const float o3 = 1.f - z[3], o4 = 1.f - z[4];
        float pLo[8];
        #pragma unroll
        for (int i = 0; i < 8; ++i)
            pLo[i] = ((i & 1) ? z[0] : o0) * ((i & 2) ? z[1] : o1) * ((i & 4) ? z[2] : o2);
        const float t5 = h ? z[5] : (1.f - z[5]);   // expert bit 5 fixed per thread
        float pHi[4];
        #pragma unroll
        for (int cc = 0; cc < 4; ++cc)
            pHi[cc] = ((cc & 1) ? z[3] : o3) * ((cc & 2) ? z[4] : o4) * t5;
        const float we = w[e] * inv;
        #pragma unroll
        for (int n = 0; n < 32; ++n)
            outv[n] = fmaf(we * pLo[n & 7], pHi[n >> 3], outv[n]);
    }

    float* og = out + (size_t)(rowBase + r) * 64 + h * 32;
    #pragma unroll
    for (int q = 0; q < 8; ++q) {
        const float4 val = make_float4(outv[q * 4 + 0], outv[q * 4 + 1],
                                       outv[q * 4 + 2], outv[q * 4 + 3]);
        *(float4*)(og + q * 4) = val;
    }
}

extern "C" void kernel_launch(void* const* d_in, const int* in_sizes, int n_in,
                              void* d_out, int out_size, void* d_ws, size_t ws_size,
                              hipStream_t stream) {
    const float* x  = (const float*)d_in[0];
    const float* Gw = (const float*)d_in[1];
    const float* Gb = (const float*)d_in[2];
    const float* Ww = (const float*)d_in[3];
    const float* Wb = (const float*)d_in[4];
    float* out = (float*)d_out;

    const int Brows = in_sizes[0] / P_DIM;                       // 65536
    const size_t shmem = (size_t)(2 * BUF_FLT) * sizeof(float);  // 104448 B

    dim3 grid(Brows / B_ROWS), block(256);
    hipLaunchKernelGGL(dselect_fused_kernel, grid, block, shmem, stream,
                       x, Gw, Gb, Ww, Wb, out);
}